// QLSTMPlus_65481071410790
// MI455X (gfx1250) — compile-verified
//
#include <hip/hip_runtime.h>
#include <math.h>

// QLSTM persistent scan for MI455X (gfx1250, wave32, WMMA).
// T=512 serial steps; B=256 independent rows -> 16 workgroups x 16 rows,
// each runs the full scan with zero inter-WGP sync. fp32 WMMA 16x16x4.
// Weight fragments live in VGPRs (weight-stationary); activations stream
// through LDS; barriers are LDS-only so global output stores never stall.

#define T_STEPS 512
#define BATCH   256
#define INDIM   128
#define HID     128
#define NQ      8
#define FEAT    256   // INDIM + HID
#define NG      32    // 4 gates * NQ angle columns
#define ROWS    16    // batch rows per workgroup
#define NBLK    (BATCH / ROWS)
#define NSPLIT  4     // K-split factor for the angle GEMM

// LDS strides (floats): both 16-lane halves of every WMMA fragment hit
// disjoint bank ranges.
#define C_STRIDE    260   // combined [x|h]  [16][256]
#define WT_STRIDE   260   // W (row-major)   [32][256]
#define WH_STRIDE   12    // W_hid           [128][8]
#define PART_STRIDE 36    // partials        [4][16][32] ; slab 0 reused for q

typedef float v2f __attribute__((ext_vector_type(2)));
typedef float v4f __attribute__((ext_vector_type(4)));
typedef float v8f __attribute__((ext_vector_type(8)));

__device__ __forceinline__ float fast_sigmoid(float x) {
    return __builtin_amdgcn_rcpf(1.0f + __expf(-x));   // v_exp + v_rcp
}
__device__ __forceinline__ float fast_tanh(float x) {
#if __has_builtin(__builtin_amdgcn_tanhf)
    return __builtin_amdgcn_tanhf(x);                  // V_TANH_F32
#else
    float e = __expf(2.0f * x);
    return 1.0f - 2.0f * __builtin_amdgcn_rcpf(e + 1.0f);
#endif
}

// Workgroup barrier that only orders LDS traffic: avoids the s_wait_storecnt 0
// drain of the streaming global output stores that __syncthreads() implies.
__device__ __forceinline__ void lds_barrier() {
#if __has_builtin(__builtin_amdgcn_s_barrier_signal) && \
    __has_builtin(__builtin_amdgcn_s_barrier_wait)   && \
    __has_builtin(__builtin_amdgcn_fence)
    __builtin_amdgcn_fence(__ATOMIC_RELEASE, "workgroup", "local");
    __builtin_amdgcn_s_barrier_signal(-1);
    __builtin_amdgcn_s_barrier_wait(-1);
    __builtin_amdgcn_fence(__ATOMIC_ACQUIRE, "workgroup", "local");
#else
    __syncthreads();
#endif
}

__global__ __launch_bounds__(256) void qlstm_scan(
    const float* __restrict__ x,      // [T, B, INDIM]
    const float* __restrict__ Wf,     // [NQ, FEAT]
    const float* __restrict__ Wi,
    const float* __restrict__ Wg,
    const float* __restrict__ Wo,
    const float* __restrict__ W_hid,  // [HID, NQ]
    const float* __restrict__ b_hid,  // [HID]
    float* __restrict__ out)          // [T*B*HID] ++ hx[B*HID] ++ cx[B*HID]
{
    __shared__ float sC   [ROWS * C_STRIDE];             // [x_t | h] per row
    __shared__ float sWT  [NG * WT_STRIDE];              // W rows: gate*8+j, k<256
    __shared__ float sWh  [HID * WH_STRIDE];             // W_hid rows: n<128, k<8
    __shared__ float sPart[NSPLIT * ROWS * PART_STRIDE]; // K-split partials; q in slab 0

    const int tid  = threadIdx.x;
    const int wave = tid >> 5;       // 0..7
    const int lane = tid & 31;
    const int hi   = lane >> 4;      // WMMA fragment half select
    const int r    = lane & 15;
    const int b0   = blockIdx.x * ROWS;

    // ---- one-time: pack weights, stage x_0, zero h ----
    for (int i = tid; i < NG * FEAT; i += 256) {
        int n = i >> 8, k = i & 255, g = n >> 3;
        const float* Wp = (g == 0) ? Wf : (g == 1) ? Wi : (g == 2) ? Wg : Wo;
        sWT[n * WT_STRIDE + k] = Wp[(n & 7) * FEAT + k];
    }
    for (int i = tid; i < HID * NQ; i += 256) {
        int n = i >> 3, k = i & 7;
        sWh[n * WH_STRIDE + k] = W_hid[i];
    }
    for (int i = tid; i < ROWS * INDIM; i += 256) {
        int row = i >> 7, col = i & 127;
        sC[row * C_STRIDE + col]         = x[((size_t)b0 + row) * INDIM + col];
        sC[row * C_STRIDE + INDIM + col] = 0.0f;   // h = 0
    }

    // phase-1 job: wave -> (K-chunk, angle N-tile)
    const int ks    = wave >> 1;            // 0..3 : 64-wide K chunk of FEAT
    const int kbase = ks * 64;
    const int n1    = (wave & 1) * 16 + r;  // angle column 0..31

    // phase-3 job: wave -> hidden N-tile
    const int n3     = wave * 16 + r;       // hidden column 0..127
    const float bias = b_hid[n3];

    // x staging job for phase 3: thread -> 8 floats of next x slice
    const int xrow8 = tid >> 4;             // 0..15
    const int xcol8 = (tid & 15) * 8;       // 0..120

    v8f c;                                  // cell state, fragment-resident
    #pragma unroll
    for (int e = 0; e < 8; ++e) c[e] = 0.0f;

    __syncthreads();

    // ---- weight-stationary: hoist all B-fragments into registers ----
    v2f wtb[16];                            // phase-1 B frags (32 floats/lane)
    #pragma unroll
    for (int ii = 0; ii < 8; ++ii) {
        const int ka = kbase + ii * 8 + 2 * hi;
        wtb[2 * ii + 0] = *(const v2f*)&sWT[n1 * WT_STRIDE + ka];
        wtb[2 * ii + 1] = *(const v2f*)&sWT[n1 * WT_STRIDE + ka + 4];
    }
    v2f whb0 = *(const v2f*)&sWh[n3 * WH_STRIDE + 2 * hi];      // phase-3 B frags
    v2f whb1 = *(const v2f*)&sWh[n3 * WH_STRIDE + 4 + 2 * hi];

    for (int t = 0; t < T_STEPS; ++t) {
        // ---- phase 1: partial[ks] = combined(16 x 64-chunk) @ W^T chunk ----
        v8f acc0, acc1;
        #pragma unroll
        for (int e = 0; e < 8; ++e) { acc0[e] = 0.0f; acc1[e] = 0.0f; }

        #pragma unroll
        for (int ii = 0; ii < 8; ++ii) {
            const int ka = kbase + ii * 8 + 2 * hi;
            v2f a0 = *(const v2f*)&sC[r * C_STRIDE + ka];
            v2f a1 = *(const v2f*)&sC[r * C_STRIDE + ka + 4];
            acc0 = __builtin_amdgcn_wmma_f32_16x16x4_f32(
                       false, a0, false, wtb[2 * ii + 0], (short)0, acc0, false, false);
            acc1 = __builtin_amdgcn_wmma_f32_16x16x4_f32(
                       false, a1, false, wtb[2 * ii + 1], (short)0, acc1, false, false);
        }
        #pragma unroll
        for (int e = 0; e < 8; ++e) {
            int m = e + 8 * hi;             // C-frag row
            sPart[(ks * ROWS + m) * PART_STRIDE + n1] = acc0[e] + acc1[e];
        }
        lds_barrier();

        // ---- phase 2: q = cumprod(cos(sum of partials)), in place in slab 0 ----
        if (tid < ROWS * 4) {
            int b = tid >> 2, g = tid & 3;
            float cp = 1.0f;
            #pragma unroll
            for (int j = 0; j < NQ; ++j) {
                int n = g * NQ + j;
                float a = sPart[(0 * ROWS + b) * PART_STRIDE + n]
                        + sPart[(1 * ROWS + b) * PART_STRIDE + n]
                        + sPart[(2 * ROWS + b) * PART_STRIDE + n]
                        + sPart[(3 * ROWS + b) * PART_STRIDE + n];
                cp *= __cosf(a);
                sPart[b * PART_STRIDE + n] = cp;   // q overwrites slab 0
            }
        }
        lds_barrier();

        // ---- phase 3: issue x_{t+1} loads early, gate GEMMs, cell update ----
        v4f xn0, xn1;
        const bool have_next = (t + 1 < T_STEPS);
        if (have_next) {                    // block-uniform branch
            const float* nx = x + ((size_t)(t + 1) * BATCH + b0 + xrow8) * INDIM + xcol8;
            xn0 = *(const v4f*)(nx);
            xn1 = *(const v4f*)(nx + 4);
            if (t + 2 < T_STEPS && tid < 64) {   // prefetch 2 steps ahead
                const char* np = (const char*)(x + ((size_t)(t + 2) * BATCH + b0) * INDIM)
                               + (size_t)tid * 128;
                __builtin_prefetch(np, 0, 1);
            }
        }

        v8f gacc[4];
        #pragma unroll
        for (int g = 0; g < 4; ++g) {
            const int ka = 2 * hi;
            v2f a0 = *(const v2f*)&sPart[r * PART_STRIDE + g * NQ + ka];
            v2f a1 = *(const v2f*)&sPart[r * PART_STRIDE + g * NQ + ka + 4];
            v8f z0, z1;
            #pragma unroll
            for (int e = 0; e < 8; ++e) { z0[e] = 0.0f; z1[e] = 0.0f; }
            z0 = __builtin_amdgcn_wmma_f32_16x16x4_f32(
                     false, a0, false, whb0, (short)0, z0, false, false);
            z1 = __builtin_amdgcn_wmma_f32_16x16x4_f32(
                     false, a1, false, whb1, (short)0, z1, false, false);
            #pragma unroll
            for (int e = 0; e < 8; ++e) gacc[g][e] = z0[e] + z1[e];
        }

        float* hout = out + ((size_t)t * BATCH + b0) * HID;
        #pragma unroll
        for (int e = 0; e < 8; ++e) {
            int m = e + 8 * hi;
            float fg = fast_sigmoid(gacc[0][e] + bias);
            float ig = fast_sigmoid(gacc[1][e] + bias);
            float gg = fast_tanh   (gacc[2][e] + bias);
            float og = fast_sigmoid(gacc[3][e] + bias);
            float cv = fg * c[e] + ig * gg;
            c[e] = cv;
            float hv = og * fast_tanh(cv);
            sC[m * C_STRIDE + INDIM + n3] = hv;   // h half of combined buffer
            hout[m * HID + n3]            = hv;   // outputs[t] (never drained mid-loop)
        }

        if (have_next) {                    // stage x_{t+1} into combined buffer
            float* dst = &sC[xrow8 * C_STRIDE + xcol8];
            *(v4f*)(dst)     = xn0;
            *(v4f*)(dst + 4) = xn1;
        }
        lds_barrier();   // sC/sPart reuse fence for next iteration
    }

    // ---- final states hx, cx ----
    float* hx = out + (size_t)T_STEPS * BATCH * HID;
    float* cx = hx + (size_t)BATCH * HID;
    #pragma unroll
    for (int e = 0; e < 8; ++e) {
        int m = e + 8 * hi;
        hx[(size_t)(b0 + m) * HID + n3] = sC[m * C_STRIDE + INDIM + n3];
        cx[(size_t)(b0 + m) * HID + n3] = c[e];
    }
}

extern "C" void kernel_launch(void* const* d_in, const int* in_sizes, int n_in,
                              void* d_out, int out_size, void* d_ws, size_t ws_size,
                              hipStream_t stream) {
    (void)in_sizes; (void)n_in; (void)d_ws; (void)ws_size; (void)out_size;
    const float* x     = (const float*)d_in[0];
    const float* Wf    = (const float*)d_in[1];
    const float* Wi    = (const float*)d_in[2];
    const float* Wg    = (const float*)d_in[3];
    const float* Wo    = (const float*)d_in[4];
    const float* W_hid = (const float*)d_in[5];
    const float* b_hid = (const float*)d_in[6];
    float* out = (float*)d_out;

    qlstm_scan<<<NBLK, 256, 0, stream>>>(x, Wf, Wi, Wg, Wo, W_hid, b_hid, out);
}